// SAGE_24026047054429
// MI455X (gfx1250) — compile-verified
//
#include <hip/hip_runtime.h>
#include <hip/hip_bf16.h>

// ---------------------------------------------------------------------------
// 3-layer GraphSAGE (mean aggr) for MI455X / gfx1250.
//   per layer: agg = scatter_mean(x[src] -> dst); out = agg@Wl + b + x_tgt@Wr
//   layers 0,1 followed by exact GELU.
// Scatter/gather is the roofline bottleneck (~1-1.5 GB HBM traffic); agg
// buffers fit in the 192MB L2 so the f32 atomics RMW in-cache. GEMMs use
// fp32 WMMA (V_WMMA_F32_16X16X4_F32) to keep reference precision.
// ---------------------------------------------------------------------------

typedef __attribute__((ext_vector_type(2))) float v2f;
typedef __attribute__((ext_vector_type(8))) float v8f;

#define CN0 1000000
#define CN1 100000
#define CN2 10000
#define CN3 1024
#define CE0 1500000
#define CE1 150000
#define CE2 15360
#define CD  128
#define CH  512

__device__ __forceinline__ void atom_add_f32(float* p, float v) {
  // lowers to global_atomic_add_f32 (no CAS loop)
  unsafeAtomicAdd(p, v);
}

// One block per edge; threads stride over the F features.
__global__ void sage_scatter(const float* __restrict__ X,
                             const int* __restrict__ src,
                             const int* __restrict__ dst,
                             float* __restrict__ Agg,
                             float* __restrict__ Cnt,
                             int E, int F) {
  int e = blockIdx.x;
  if (e >= E) return;
  int s = src[e];
  int d = dst[e];
  const float* xs = X + (size_t)s * F;
  float* ag = Agg + (size_t)d * F;
  for (int f = threadIdx.x; f < F; f += blockDim.x)
    atom_add_f32(&ag[f], xs[f]);
  if (threadIdx.x == 0) atom_add_f32(&Cnt[d], 1.0f);
}

// Agg[i] /= max(cnt(row), 1)
__global__ void sage_mean(float* __restrict__ Agg,
                          const float* __restrict__ Cnt,
                          int total, int F) {
  int i = blockIdx.x * blockDim.x + threadIdx.x;
  if (i >= total) return;
  float c = Cnt[i / F];
  Agg[i] *= 1.0f / fmaxf(c, 1.0f);
}

// Out[N,Hout] = Agg@Wl + Xt@Wr + bias, optional exact GELU.
// grid = (N/16, Hout/64), block = 128 (4 waves); each wave owns a 16x16 tile.
// fp32 WMMA 16x16x4. Per ISA 7.12.2:
//   A lane L, vgpr v : element (M = L%16, K = kbase + 2*(L/16) + v)  -> float2
//   B lane L, vgpr v : element (K = kbase + 2*(L/16) + v, N = L%16)
//   C lane L, vgpr r : element (M = r + 8*(L/16), N = L%16)
__global__ void sage_gemm_wmma(const float* __restrict__ Agg,
                               const float* __restrict__ Xt,
                               const float* __restrict__ Wl,
                               const float* __restrict__ Wr,
                               const float* __restrict__ bias,
                               float* __restrict__ Out,
                               int K, int Hout, int do_gelu) {
  const int wave = threadIdx.x >> 5;
  const int lane = threadIdx.x & 31;
  const int half = lane >> 4;    // 0 or 1
  const int lr   = lane & 15;
  const int m0   = blockIdx.x << 4;
  const int n0   = (blockIdx.y << 6) + (wave << 4);
  const int arow = m0 + lr;
  const int bcol = n0 + lr;

  v8f c = {};

  const float* arow_p = Agg + (size_t)arow * K;
#pragma unroll 4
  for (int k = 0; k < K; k += 4) {
    const int kb = k + 2 * half;
    v2f a = *(const v2f*)(arow_p + kb);
    v2f b;
    b.x = Wl[(size_t)kb * Hout + bcol];
    b.y = Wl[(size_t)(kb + 1) * Hout + bcol];
    c = __builtin_amdgcn_wmma_f32_16x16x4_f32(false, a, false, b,
                                              (short)0, c, false, false);
  }

  const float* xrow_p = Xt + (size_t)arow * K;
#pragma unroll 4
  for (int k = 0; k < K; k += 4) {
    const int kb = k + 2 * half;
    v2f a = *(const v2f*)(xrow_p + kb);
    v2f b;
    b.x = Wr[(size_t)kb * Hout + bcol];
    b.y = Wr[(size_t)(kb + 1) * Hout + bcol];
    c = __builtin_amdgcn_wmma_f32_16x16x4_f32(false, a, false, b,
                                              (short)0, c, false, false);
  }

  const float bv = bias[bcol];
#pragma unroll
  for (int r = 0; r < 8; ++r) {
    float v = c[r] + bv;
    if (do_gelu) v = 0.5f * v * (1.0f + erff(v * 0.70710678118654752f));
    Out[(size_t)(m0 + r + 8 * half) * Hout + bcol] = v;
  }
}

static inline size_t align256(size_t x) { return (x + 255) & ~(size_t)255; }

extern "C" void kernel_launch(void* const* d_in, const int* in_sizes, int n_in,
                              void* d_out, int out_size, void* d_ws, size_t ws_size,
                              hipStream_t stream) {
  (void)in_sizes; (void)n_in; (void)out_size; (void)ws_size;

  const float* x       = (const float*)d_in[0];
  const int*   ei0_src = (const int*)d_in[1];
  const int*   ei0_dst = (const int*)d_in[2];
  const int*   ei1_src = (const int*)d_in[3];
  const int*   ei1_dst = (const int*)d_in[4];
  const int*   ei2_src = (const int*)d_in[5];
  const int*   ei2_dst = (const int*)d_in[6];
  const float* W_l0    = (const float*)d_in[7];
  const float* b_l0    = (const float*)d_in[8];
  const float* W_r0    = (const float*)d_in[9];
  const float* W_l1    = (const float*)d_in[10];
  const float* b_l1    = (const float*)d_in[11];
  const float* W_r1    = (const float*)d_in[12];
  const float* W_l2    = (const float*)d_in[13];
  const float* b_l2    = (const float*)d_in[14];
  const float* W_r2    = (const float*)d_in[15];
  float* out = (float*)d_out;

  // Workspace layout (agg buffer reused across layers; max size is layer 0):
  char* ws = (char*)d_ws;
  size_t off = 0;
  float* agg = (float*)(ws + off); off += align256((size_t)CN1 * CD * sizeof(float));
  float* cnt = (float*)(ws + off); off += align256((size_t)CN1 * sizeof(float));
  float* h1  = (float*)(ws + off); off += align256((size_t)CN1 * CH * sizeof(float));
  float* h2  = (float*)(ws + off); off += align256((size_t)CN2 * CH * sizeof(float));

  // ---------------- Layer 0: (N0,D) -> (N1,H), GELU ----------------
  hipMemsetAsync(agg, 0, (size_t)CN1 * CD * sizeof(float), stream);
  hipMemsetAsync(cnt, 0, (size_t)CN1 * sizeof(float), stream);
  sage_scatter<<<CE0, 128, 0, stream>>>(x, ei0_src, ei0_dst, agg, cnt, CE0, CD);
  {
    int total = CN1 * CD;
    sage_mean<<<(total + 255) / 256, 256, 0, stream>>>(agg, cnt, total, CD);
  }
  sage_gemm_wmma<<<dim3(CN1 / 16, CH / 64), 128, 0, stream>>>(
      agg, x, W_l0, W_r0, b_l0, h1, CD, CH, 1);

  // ---------------- Layer 1: (N1,H) -> (N2,H), GELU ----------------
  hipMemsetAsync(agg, 0, (size_t)CN2 * CH * sizeof(float), stream);
  hipMemsetAsync(cnt, 0, (size_t)CN2 * sizeof(float), stream);
  sage_scatter<<<CE1, 128, 0, stream>>>(h1, ei1_src, ei1_dst, agg, cnt, CE1, CH);
  {
    int total = CN2 * CH;
    sage_mean<<<(total + 255) / 256, 256, 0, stream>>>(agg, cnt, total, CH);
  }
  sage_gemm_wmma<<<dim3(CN2 / 16, CH / 64), 128, 0, stream>>>(
      agg, h1, W_l1, W_r1, b_l1, h2, CH, CH, 1);

  // ---------------- Layer 2: (N2,H) -> (N3,D), no GELU ----------------
  hipMemsetAsync(agg, 0, (size_t)CN3 * CH * sizeof(float), stream);
  hipMemsetAsync(cnt, 0, (size_t)CN3 * sizeof(float), stream);
  sage_scatter<<<CE2, 128, 0, stream>>>(h2, ei2_src, ei2_dst, agg, cnt, CE2, CH);
  {
    int total = CN3 * CH;
    sage_mean<<<(total + 255) / 256, 256, 0, stream>>>(agg, cnt, total, CH);
  }
  sage_gemm_wmma<<<dim3(CN3 / 16, CD / 64), 128, 0, stream>>>(
      agg, h2, W_l2, W_r2, b_l2, out, CH, CD, 0);
}